// DotProductAttention_G_35854386987228
// MI455X (gfx1250) — compile-verified
//
#include <hip/hip_runtime.h>
#include <hip/hip_bf16.h>
#include <math.h>

// Problem constants (match reference)
#define BATCH 32
#define SEQ   1024
#define DIM   64
#define MASK_VALUE -1000000.0f
#define SCALE 0.125f            // 1/sqrt(64)

// LDS score strip: 16 rows x SEQ cols, padded row stride to break bank alignment
#define SSTRIDE 1032            // 1032 % 64 = 8 -> rotates banks per row

typedef __attribute__((ext_vector_type(2))) float v2f;
typedef __attribute__((ext_vector_type(8))) float v8f;

__launch_bounds__(256)
__global__ void attn_fp32_wmma_kernel(const float* __restrict__ q,
                                      const float* __restrict__ k,
                                      const float* __restrict__ v,
                                      const int*   __restrict__ valid_lens,
                                      float*       __restrict__ out) {
    __shared__ float sS[16 * SSTRIDE];   // score / prob strip (16 x 1024, padded)
    __shared__ float sO[16 * DIM];       // cross-wave O reduction tile
    __shared__ float sMax[16];
    __shared__ float sInv[16];           // reciprocal of softmax row sum

    const int b    = blockIdx.y;
    const int q0   = blockIdx.x * 16;                 // first query row of this tile
    const int tid  = threadIdx.x;
    const int lane = tid & 31;
    const int wave = tid >> 5;                        // 0..7
    const int m16  = lane & 15;                       // row (A) / col (B,C,D) index
    const int hi   = lane >> 4;                       // 0 or 1 (lane half)
    const int khalf = hi << 1;                        // K offset 0 or 2 within a K=4 chunk

    const int vlen = valid_lens[b];

    // ---- Phase 0: zero the O-reduction tile (1024 floats, 256 threads) ----
    #pragma unroll
    for (int i = 0; i < 4; ++i) sO[tid + 256 * i] = 0.0f;

    // ---- Load Q A-fragments for the whole tile: 16 chunks of K=4 -> 32 floats/lane ----
    // A layout (ISA 7.12.2, 32-bit A 16x4): lane l holds A[m=l&15][4c+2*(l>>4)] .. +1
    const float* Qb = q + ((size_t)b * SEQ + q0) * DIM + (size_t)m16 * DIM + khalf;
    v2f qa[16];
    #pragma unroll
    for (int c = 0; c < 16; ++c)
        qa[c] = *(const v2f*)(Qb + 4 * c);

    // ---- Phase 1: S = (Q K^T) * scale, masked, into LDS. Wave w owns key tiles w,w+8,... ----
    const float* Kb = k + (size_t)b * SEQ * DIM;
    for (int jj = 0; jj < 8; ++jj) {
        const int j = wave + 8 * jj;                  // key tile index, uniform per wave
        v8f acc = {};
        // B layout (4x16 f32): lane l holds B[2*(l>>4)][n=l&15], B[2*(l>>4)+1][n]
        // B[kk][n] = keys[16j+n][4c+kk]  (K^T), so a float2 from keys row 16j+n
        const float* Kt = Kb + (size_t)(16 * j + m16) * DIM + khalf;
        #pragma unroll
        for (int c = 0; c < 16; ++c) {
            v2f kb = *(const v2f*)(Kt + 4 * c);
            acc = __builtin_amdgcn_wmma_f32_16x16x4_f32(
                      false, qa[c], false, kb, (short)0, acc, false, false);
        }
        const int col = 16 * j + m16;                 // key (column) index of this lane
        const bool masked = (col >= vlen);
        #pragma unroll
        for (int r = 0; r < 8; ++r) {
            const int m = r + (hi << 3);              // C/D layout: row = r + 8*half
            float s = acc[r] * SCALE;
            if (masked) s = MASK_VALUE;
            sS[m * SSTRIDE + col] = s;
        }
    }
    __syncthreads();

    // ---- Phase 2a: row max (16 threads per row, strided cols, 16-lane shuffle tree) ----
    const int row = tid >> 4;                         // 0..15
    const int sub = tid & 15;
    float pmax = -INFINITY;
    for (int i = 0; i < SEQ / 16; ++i)
        pmax = fmaxf(pmax, sS[row * SSTRIDE + sub + 16 * i]);
    #pragma unroll
    for (int off = 8; off >= 1; off >>= 1)
        pmax = fmaxf(pmax, __shfl_xor(pmax, off, 32));
    if (sub == 0) sMax[row] = pmax;
    __syncthreads();

    // ---- Phase 2b: exp(s - max) written back to LDS + row sum (store reciprocal) ----
    const float rmax = sMax[row];
    float psum = 0.0f;
    for (int i = 0; i < SEQ / 16; ++i) {
        const int idx = row * SSTRIDE + sub + 16 * i;
        const float e = __expf(sS[idx] - rmax);       // masked entries underflow to 0
        sS[idx] = e;
        psum += e;
    }
    #pragma unroll
    for (int off = 8; off >= 1; off >>= 1)
        psum += __shfl_xor(psum, off, 32);
    if (sub == 0) sInv[row] = 1.0f / psum;            // psum >= 1 (max elem -> exp(0))
    __syncthreads();

    // ---- Phase 3: write normalized attn_w (coalesced, non-temporal: write-once stream) ----
    // d_out layout: [attn_score (B,L,D)] then [attn_w (B,L,L)]
    float* aw = out + (size_t)BATCH * SEQ * DIM + ((size_t)b * SEQ + q0) * SEQ;
    for (int i = 0; i < (16 * SEQ) / 256; ++i) {
        const int t  = tid + 256 * i;
        const int r2 = t >> 10;                       // /1024
        const int c2 = t & (SEQ - 1);
        __builtin_nontemporal_store(sS[r2 * SSTRIDE + c2] * sInv[r2],
                                    &aw[(size_t)r2 * SEQ + c2]);
    }

    // ---- Phase 4: O_unnorm = P_unnorm * V via fp32 WMMA; wave-partial then LDS atomics ----
    const float* Vb = v + (size_t)b * SEQ * DIM;
    v8f oacc[4] = {};                                 // 4 column tiles of D=64
    for (int jj = 0; jj < 8; ++jj) {
        const int j = wave + 8 * jj;                  // key tile (K-dim slice), uniform per wave
        #pragma unroll
        for (int c2 = 0; c2 < 4; ++c2) {              // K=4 chunks within the 16-key slice
            const int kb0 = 16 * j + 4 * c2 + khalf;  // K row of this lane's fragment
            // A-frag of P from LDS: lane l -> P[m=l&15][kb0], P[m][kb0+1]
            const v2f pa = *(const v2f*)&sS[m16 * SSTRIDE + kb0];
            #pragma unroll
            for (int t = 0; t < 4; ++t) {             // D column tiles
                const float* pv = Vb + (size_t)kb0 * DIM + 16 * t + m16;
                v2f vb;
                vb.x = pv[0];                         // V[kb0][16t+n]
                vb.y = pv[DIM];                       // V[kb0+1][16t+n]
                oacc[t] = __builtin_amdgcn_wmma_f32_16x16x4_f32(
                              false, pa, false, vb, (short)0, oacc[t], false, false);
            }
        }
    }
    #pragma unroll
    for (int t = 0; t < 4; ++t)
        #pragma unroll
        for (int r = 0; r < 8; ++r) {
            const int m = r + (hi << 3);
            atomicAdd(&sO[m * DIM + 16 * t + m16], oacc[t][r]);   // ds_add_f32
        }
    __syncthreads();

    // ---- Phase 5: normalize and write attn_score (non-temporal) ----
    float* os = out + ((size_t)b * SEQ + q0) * DIM;
    #pragma unroll
    for (int i = 0; i < (16 * DIM) / 256; ++i) {
        const int t  = tid + 256 * i;
        const int r2 = t >> 6;                        // /64
        const int c2 = t & (DIM - 1);
        __builtin_nontemporal_store(sO[t] * sInv[r2], &os[(size_t)r2 * DIM + c2]);
    }
}

extern "C" void kernel_launch(void* const* d_in, const int* in_sizes, int n_in,
                              void* d_out, int out_size, void* d_ws, size_t ws_size,
                              hipStream_t stream) {
    (void)in_sizes; (void)n_in; (void)out_size; (void)d_ws; (void)ws_size;
    const float* q  = (const float*)d_in[0];
    const float* k  = (const float*)d_in[1];
    const float* v  = (const float*)d_in[2];
    const int* vlen = (const int*)d_in[3];
    float* out = (float*)d_out;

    dim3 grid(SEQ / 16, BATCH);   // (query tile, batch)
    attn_fp32_wmma_kernel<<<grid, 256, 0, stream>>>(q, k, v, vlen, out);
}